// NNet_2d_61589831024790
// MI455X (gfx1250) — compile-verified
//
#include <hip/hip_runtime.h>

#define CLASS_NUM 4
#define NUM_MICRO 4
#define IN_PLANES 16
#define BSIZE 16
#define HH 512
#define WW 512
#define HWSZ (HH * WW)              // 262144
#define NPIX (BSIZE * HWSZ)         // 4194304
#define PXBLK 128                   // pixels per wave-iteration (32 quads -> 32 WMMAs)
#define NBLK (NPIX / PXBLK)         // 32768 blocks
#define NGRP (CLASS_NUM * NUM_MICRO)  // 16
#define MOM 0.99f

typedef float v2f __attribute__((ext_vector_type(2)));
typedef float v8f __attribute__((ext_vector_type(8)));

// ---------------- workspace zeroing (d_ws is poisoned by harness) ------------
__global__ void zero_ws_kernel(float* ws) {
    int t = threadIdx.x;
    if (t < 256 + NGRP) ws[t] = 0.0f;
}

// ---------------- main segmented reduction as one-hot GEMM via WMMA ----------
// sums[g*16+c] += rep[pix,c] for every valid pixel with gid==g ; counts[g] += 1
__global__ void proto_segreduce_kernel(const float* __restrict__ rep,
                                       const int* __restrict__ pmi,      // [NPIX*4]
                                       const int* __restrict__ target,   // [NPIX]
                                       const int* __restrict__ smask,    // [HWSZ]
                                       const unsigned char* __restrict__ cond, // [NPIX] bool
                                       float* __restrict__ gsums,        // [256]
                                       float* __restrict__ gcounts)      // [16]
{
    __shared__ float lsums[256];
    __shared__ float lcounts[NGRP];

    const int tid = threadIdx.x;
    lsums[tid] = 0.0f;
    if (tid < NGRP) lcounts[tid] = 0.0f;
    __syncthreads();

    const int lane    = tid & 31;
    const unsigned laneM = lane & 15;
    const bool hiHalf = lane >= 16;
    const int sh0     = hiHalf ? 16 : 0;   // byte position of this half's first K pixel
    const int sh1     = sh0 + 8;
    const int d0      = hiHalf ? 2 : 0;    // pixel offset within quad for this half

    const int waveId = tid >> 5;
    const int wavesPerBlk = blockDim.x >> 5;
    const int gw = blockIdx.x * wavesPerBlk + waveId;   // uniform within wave
    const int nw = gridDim.x * wavesPerBlk;

    v8f accE = {};         // even quads
    v8f accO = {};         // odd quads (breaks WMMA C->D dependency chain)
    v2f cntV = {};         // packed per-lane count partials (one v_pk_add_f32 per WMMA)

    for (int blk = gw; blk < NBLK; blk += nw) {          // uniform cond -> EXEC all 1s
        const int pix0 = blk << 7;                       // 128-pixel block start
        const int bs   = pix0 >> 18;                     // / HWSZ (never crosses bs)
        const int hw0  = pix0 & (HWSZ - 1);

        // ---- meta phase: lane l computes byte-packed gids of quad l (pixels 4l..4l+3)
        const int q = pix0 + (lane << 2);
        const int4 t4 = *(const int4*)(target + q);
        const unsigned c4 = *(const unsigned int*)(cond + q);   // 4 bool bytes
        const int4 s4 = *(const int4*)(smask + (hw0 + (lane << 2)));
        const int m0 = pmi[((q + 0) << 2) + t4.x];
        const int m1 = pmi[((q + 1) << 2) + t4.y];
        const int m2 = pmi[((q + 2) << 2) + t4.z];
        const int m3 = pmi[((q + 3) << 2) + t4.w];

        const unsigned g0 = ((s4.x == 1) & (t4.x != 0) & ((c4 & 0x000000ffu) != 0))
                                ? (unsigned)((t4.x << 2) + m0) : 255u;
        const unsigned g1 = ((s4.y == 1) & (t4.y != 0) & ((c4 & 0x0000ff00u) != 0))
                                ? (unsigned)((t4.y << 2) + m1) : 255u;
        const unsigned g2 = ((s4.z == 1) & (t4.z != 0) & ((c4 & 0x00ff0000u) != 0))
                                ? (unsigned)((t4.z << 2) + m2) : 255u;
        const unsigned g3 = ((s4.w == 1) & (t4.w != 0) & ((c4 & 0xff000000u) != 0))
                                ? (unsigned)((t4.w << 2) + m3) : 255u;
        const unsigned pk = g0 | (g1 << 8) | (g2 << 16) | (g3 << 24);

        // per-lane rep pointer: channel-plane laneM of image bs, at this half's K offset
        const float* lrep = rep + (((long long)(bs * IN_PLANES + (int)laneM)) << 18)
                                + hw0 + d0;

#pragma unroll
        for (int j = 0; j < 32; ++j) {   // 32 quads = 32 WMMAs per 128-pixel block
            // one readlane per WMMA: packed gids of quad j live in lane j
            const unsigned pj = (unsigned)__builtin_amdgcn_readlane((int)pk, j);
            const unsigned e0 = (pj >> sh0) & 0xffu;   // gid of this half's pixel K0/K2
            const unsigned e1 = (pj >> sh1) & 0xffu;   // gid of this half's pixel K1/K3
            const float a0 = (e0 == laneM) ? 1.0f : 0.0f;
            const float a1 = (e1 == laneM) ? 1.0f : 0.0f;
            v2f A = {a0, a1};

            // B (4x16): lane reads its channel at this half's two pixels: one b64 NT load
            v2f B = __builtin_nontemporal_load((const v2f*)(lrep + 4 * j));

            cntV += A;   // counts fall out of A: single packed f32 add

            if (j & 1)
                accO = __builtin_amdgcn_wmma_f32_16x16x4_f32(
                    false, A, false, B, (short)0, accO, false, false);
            else
                accE = __builtin_amdgcn_wmma_f32_16x16x4_f32(
                    false, A, false, B, (short)0, accE, false, false);
        }
    }

    const v8f acc = accE + accO;          // merge the two chains (packed f32 adds)
    const float cntAcc = cntV.x + cntV.y; // per-lane partial of counts[laneM]

    // ---- wave accumulator -> LDS (C/D layout: vgpr r = rows r / r+8) ----
#pragma unroll
    for (int r = 0; r < 8; ++r) {
        const int row = hiHalf ? (r + 8) : r;
        atomicAdd(&lsums[row * 16 + (int)laneM], acc[r]);
    }
    atomicAdd(&lcounts[laneM], cntAcc);
    __syncthreads();

    // ---- one global atomic set per workgroup ----
    atomicAdd(&gsums[tid], lsums[tid]);
    if (tid < NGRP) atomicAdd(&gcounts[tid], lcounts[tid]);
}

// ---------------- EMA + L2 normalize (256 values total) ----------------------
__global__ void proto_finalize_kernel(const float* __restrict__ protos,  // [256]
                                      const float* __restrict__ gsums,   // [256]
                                      const float* __restrict__ gcounts, // [16]
                                      float* __restrict__ out)           // [256]
{
    __shared__ float usq[256];
    const int t = threadIdx.x;        // t = g*16 + c
    const int g = t >> 4;

    const float cnt  = gcounts[g];
    const float mean = gsums[t] / fmaxf(cnt, 1.0f);
    const float p    = protos[t];
    const float val  = (cnt > 0.0f) ? (MOM * p + (1.0f - MOM) * mean) : p;

    usq[t] = val * val;
    __syncthreads();

    float ss = 0.0f;
#pragma unroll
    for (int c = 0; c < 16; ++c) ss += usq[(g << 4) + c];
    const float nrm = sqrtf(ss);
    out[t] = val / fmaxf(nrm, 1e-12f);
}

// ---------------------------------------------------------------------------
extern "C" void kernel_launch(void* const* d_in, const int* in_sizes, int n_in,
                              void* d_out, int out_size, void* d_ws, size_t ws_size,
                              hipStream_t stream) {
    (void)in_sizes; (void)n_in; (void)out_size; (void)ws_size;

    const float*         rep    = (const float*)d_in[0];          // [16,16,512,512]
    const float*         protos = (const float*)d_in[1];          // [4,4,16]
    const int*           pmi    = (const int*)d_in[2];            // [16,512,512,4]
    const int*           target = (const int*)d_in[3];            // [16,512,512]
    const int*           smask  = (const int*)d_in[4];            // [512,512]
    const unsigned char* cond   = (const unsigned char*)d_in[5];  // [16,512,512] bool

    float* gsums   = (float*)d_ws;        // 256 floats
    float* gcounts = gsums + 256;         // 16 floats
    float* out     = (float*)d_out;       // 256 floats

    zero_ws_kernel<<<1, 288, 0, stream>>>(gsums);
    // 512 WGs x 256 threads = 4096 waves; 32768 blocks -> 8 iterations/wave
    proto_segreduce_kernel<<<512, 256, 0, stream>>>(rep, pmi, target, smask, cond,
                                                    gsums, gcounts);
    proto_finalize_kernel<<<1, 256, 0, stream>>>(protos, gsums, gcounts, out);
}